// Tabulated_specific_4647154614864
// MI455X (gfx1250) — compile-verified
//
#include <hip/hip_runtime.h>

#define N_ATOMS 100000
#define N_PAIRS 6400000
#define N_KNOTS 512
#define NSEG    (N_KNOTS - 1)   // 511 segments
#define NTAB    2
#define COEF_FLOATS (NTAB * 4 * NSEG)   // 4088 floats = 16352 B
#define BLOCK   256
#define PPT     2                        // pairs per thread
#define PAIRS_PER_BLOCK (BLOCK * PPT)

static_assert(N_PAIRS % PAIRS_PER_BLOCK == 0, "no tail: grid divides pairs exactly");

// ---- CDNA5 async global->LDS staging (ASYNCcnt path), guarded -------------
#if defined(__has_builtin)
#if __has_builtin(__builtin_amdgcn_global_load_async_to_lds_b32) && \
    __has_builtin(__builtin_amdgcn_s_wait_asynccnt)
#define USE_ASYNC_LDS 1
#endif
#endif

typedef __attribute__((address_space(1))) int GInt;   // global (AS1)
typedef __attribute__((address_space(3))) int LInt;   // LDS    (AS3)

// 16B vector with alignment forced down to 4; the backend emits one wide
// (b96/b128) global load per position row (verified: vmem count is invariant
// to lane-liveness forcing, so rows are already single wide loads).
typedef float f4u __attribute__((ext_vector_type(4), aligned(4)));

__global__ void zero_kernel(float* __restrict__ out, int n) {
  int i = blockIdx.x * blockDim.x + threadIdx.x;
  if (i < n) out[i] = 0.0f;
}

__device__ __forceinline__ void atomic_add_f32(float* p, float v) {
  // relaxed, device scope -> non-returning global_atomic_add_f32
  __hip_atomic_fetch_add(p, v, __ATOMIC_RELAXED, __HIP_MEMORY_SCOPE_AGENT);
}

__launch_bounds__(BLOCK, 8)
__global__ void pair_force_kernel(const float* __restrict__ q,
                                  const float* __restrict__ cell,
                                  const int*   __restrict__ z,
                                  const int*   __restrict__ nbr,
                                  const float* __restrict__ knots,
                                  const float* __restrict__ coeffs,
                                  const int*   __restrict__ inter,
                                  float*       __restrict__ force) {
  __shared__ __align__(16) float s_coef[COEF_FLOATS];
  __shared__ __align__(16) float s_knots[N_KNOTS];

  const int tid = threadIdx.x;

  // ---- stage spline tables into LDS (18.4 KB per block) ----
#ifdef USE_ASYNC_LDS
  for (int idx = tid; idx < COEF_FLOATS; idx += BLOCK) {
    __builtin_amdgcn_global_load_async_to_lds_b32(
        (GInt*)(void*)(coeffs + idx),
        (LInt*)(s_coef + idx), 0, 0);
  }
  for (int idx = tid; idx < N_KNOTS; idx += BLOCK) {
    __builtin_amdgcn_global_load_async_to_lds_b32(
        (GInt*)(void*)(knots + idx),
        (LInt*)(s_knots + idx), 0, 0);
  }
  __builtin_amdgcn_s_wait_asynccnt(0);
#else
  for (int idx = tid; idx < COEF_FLOATS; idx += BLOCK) s_coef[idx]  = coeffs[idx];
  for (int idx = tid; idx < N_KNOTS;    idx += BLOCK) s_knots[idx] = knots[idx];
#endif
  __syncthreads();

  // uniform (per-thread-invariant) parameters
  const float cx = cell[0], cy = cell[1], cz = cell[2];
  const float icx = __builtin_amdgcn_rcpf(cx);   // feeds rintf only: 1-ulp safe
  const float icy = __builtin_amdgcn_rcpf(cy);
  const float icz = __builtin_amdgcn_rcpf(cz);
  const float x0 = s_knots[0];
  const float inv_h = (float)NSEG / (s_knots[N_KNOTS - 1] - x0);
  const int i00 = inter[0], i01 = inter[1];
  const int i10 = inter[2], i11 = inter[3];

  const int base = blockIdx.x * PAIRS_PER_BLOCK + tid;

#pragma unroll
  for (int u = 0; u < PPT; ++u) {
    const int p = base + u * BLOCK;                 // no tail: grid is exact

    const int2 ij = ((const int2* __restrict__)nbr)[p];
    const int i = ij.x;
    const int j = ij.y;

    // one wide global load per position row (4B overread stays in-page)
    const f4u qi = *(const f4u*)(q + 3 * i);
    const f4u qj = *(const f4u*)(q + 3 * j);

    float dx = qi.x - qj.x;
    float dy = qi.y - qj.y;
    float dz = qi.z - qj.z;

    // minimum image; rintf == round-half-even == jnp.round
    dx -= cx * rintf(dx * icx);
    dy -= cy * rintf(dy * icy);
    dz -= cz * rintf(dz * icz);

    const float r2 = fmaf(dx, dx, fmaf(dy, dy, dz * dz));
    const float r  = sqrtf(r2);
    const float inv_r = 1.0f / r;                  // exact: scales the force

    // uniform knot grid: analytic searchsorted
    int seg = (int)floorf((r - x0) * inv_h);
    seg = seg < 0 ? 0 : (seg > NSEG - 1 ? NSEG - 1 : seg);
    const float t = r - s_knots[seg];

    const int zi = z[i];
    const int zj = z[j];

    const bool m0 = ((zi == i00) | (zi == i01)) & ((zj == i00) | (zj == i01));
    const bool m1 = ((zi == i10) | (zi == i11)) & ((zj == i10) | (zj == i11));

    float f = 0.0f;
    {
      const float a = s_coef[seg];
      const float b = s_coef[NSEG + seg];
      const float c = s_coef[2 * NSEG + seg];
      const float d = s_coef[3 * NSEG + seg];
      const float fm = fmaf(fmaf(fmaf(d, t, c), t, b), t, a);
      f += m0 ? fm : 0.0f;
    }
    {
      const float* ck = s_coef + 4 * NSEG;
      const float a = ck[seg];
      const float b = ck[NSEG + seg];
      const float c = ck[2 * NSEG + seg];
      const float d = ck[3 * NSEG + seg];
      const float fm = fmaf(fmaf(fmaf(d, t, c), t, b), t, a);
      f += m1 ? fm : 0.0f;
    }

    if (m0 | m1) {
      const float s = f * inv_r;
      const float fx = s * dx, fy = s * dy, fz = s * dz;
      float* fi = force + 3 * i;
      float* fj = force + 3 * j;
      atomic_add_f32(fi + 0,  fx);
      atomic_add_f32(fi + 1,  fy);
      atomic_add_f32(fi + 2,  fz);
      atomic_add_f32(fj + 0, -fx);
      atomic_add_f32(fj + 1, -fy);
      atomic_add_f32(fj + 2, -fz);
    }
  }
}

extern "C" void kernel_launch(void* const* d_in, const int* in_sizes, int n_in,
                              void* d_out, int out_size, void* d_ws, size_t ws_size,
                              hipStream_t stream) {
  (void)in_sizes; (void)n_in; (void)d_ws; (void)ws_size;
  const float* q      = (const float*)d_in[0];
  const float* cell   = (const float*)d_in[1];
  const int*   z      = (const int*)d_in[2];
  const int*   nbr    = (const int*)d_in[3];
  const float* knots  = (const float*)d_in[4];
  const float* coeffs = (const float*)d_in[5];
  const int*   inter  = (const int*)d_in[6];
  float*       force  = (float*)d_out;

  // zero the accumulator (harness poisons d_out)
  zero_kernel<<<(out_size + 255) / 256, 256, 0, stream>>>(force, out_size);

  const int grid = N_PAIRS / PAIRS_PER_BLOCK;   // 12500, exact
  pair_force_kernel<<<grid, BLOCK, 0, stream>>>(q, cell, z, nbr, knots, coeffs,
                                                inter, force);
}